// MultiLayerNetwork_59244778881371
// MI455X (gfx1250) — compile-verified
//
#include <hip/hip_runtime.h>
#include <math.h>

typedef __attribute__((ext_vector_type(2))) float v2f;
typedef __attribute__((ext_vector_type(4))) float v4f;
typedef __attribute__((ext_vector_type(8))) float v8f;

#define IN_N   4096
#define HID_N  8192
#define OUT_N  4096
#define TOTAL  (HID_N + OUT_N)
#define STEPS  100

// ---------------------------------------------------------------------------
// GEMV via V_WMMA_F32_16X16X4_F32:
//   y[m] = sign(t)*log1p(gain*|t| + 1e-6),  t = sum_k (W[m,k]-fb[m,k]) * x[k]
//
// One workgroup (256 thr = 8 waves, wave32) owns 16 rows; wave w reduces the
// K-slice [w*K/8, (w+1)*K/8). x is broadcast across all 16 WMMA columns so
// every accumulator column equals the row dot product.
//
// Per iteration each lane loads contiguous float4 of W, fb, x at
// kb + (lane>>4)*4 and feeds TWO WMMAs:
//   wmma0 slots: lanes<16 k={kb,kb+1}, lanes>=16 k={kb+4,kb+5}  -> a.xy/b.xy
//   wmma1 slots: lanes<16 k={kb+2,kb+3}, lanes>=16 k={kb+6,kb+7}-> a.zw/b.zw
// A and B use the same k->slot permutation, so the contraction is exact.
// Two accumulators break the WMMA dependency chain for overlap/unrolling.
// ---------------------------------------------------------------------------
__global__ __launch_bounds__(256) void gemv_wmma_kernel(
    const float* __restrict__ W, const float* __restrict__ fb,
    const float* __restrict__ x, const float* __restrict__ gain_p,
    float* __restrict__ y, int M, int K)
{
    __shared__ float partial[8][16];

    const int wave  = threadIdx.x >> 5;
    const int lane  = threadIdx.x & 31;
    const int row0  = blockIdx.x * 16;
    const int r     = lane & 15;        // row within 16-row tile
    const int khalf = lane >> 4;        // which half of the 8-wide k packet

    const int Kc   = K >> 3;            // per-wave K slice (multiple of 8)
    const int kbeg = wave * Kc;
    const int kend = kbeg + Kc;

    const float* __restrict__ wrow = W  + (size_t)(row0 + r) * (size_t)K;
    const float* __restrict__ frow = fb + (size_t)(row0 + r) * (size_t)K;

    v8f acc0 = {};
    v8f acc1 = {};
    #pragma unroll 2
    for (int kb = kbeg; kb < kend; kb += 8) {
        const int kk = kb + khalf * 4;
        // 16B contiguous streams per lane; lane-pair (r, r+16) covers the row.
        v4f aw = *(const v4f*)(wrow + kk);
        v4f af = *(const v4f*)(frow + kk);
        v4f a  = aw - af;
        v4f b  = *(const v4f*)(x + kk);   // broadcast over the 16 columns

        v2f a0 = {a.x, a.y}, a1 = {a.z, a.w};
        v2f b0 = {b.x, b.y}, b1 = {b.z, b.w};

        acc0 = __builtin_amdgcn_wmma_f32_16x16x4_f32(
                   false, a0, false, b0, (short)0, acc0, false, false);
        acc1 = __builtin_amdgcn_wmma_f32_16x16x4_f32(
                   false, a1, false, b1, (short)0, acc1, false, false);
    }
    v8f acc = acc0 + acc1;

    // Every column of C equals the row partial; C layout: VGPR j holds
    // (M=j, lanes 0-15) and (M=j+8, lanes 16-31). Take lanes 0 and 16.
    if (lane == 0) {
        #pragma unroll
        for (int j = 0; j < 8; ++j) partial[wave][j] = acc[j];
    } else if (lane == 16) {
        #pragma unroll
        for (int j = 0; j < 8; ++j) partial[wave][8 + j] = acc[j];
    }
    __syncthreads();

    // Cross-wave reduction + adaptive log compression.
    if (threadIdx.x < 16) {
        float t = 0.f;
        #pragma unroll
        for (int w2 = 0; w2 < 8; ++w2) t += partial[w2][threadIdx.x];
        const float g = gain_p[0];
        const float s = (t > 0.f) ? 1.f : ((t < 0.f) ? -1.f : 0.f);
        y[row0 + threadIdx.x] = s * log1pf(g * fabsf(t) + 1e-6f);
    }
}

// ---------------------------------------------------------------------------
// Izhikevich scan: one thread per neuron, 100 steps, writes v/u trajectories.
// v_vec/u_vec are [STEPS, TOTAL] row-major (coalesced per-step stores).
// ---------------------------------------------------------------------------
__global__ __launch_bounds__(256) void izhikevich_kernel(
    const float* __restrict__ h, const float* __restrict__ o,
    float* __restrict__ v_vec, float* __restrict__ u_vec)
{
    const int i = blockIdx.x * blockDim.x + threadIdx.x;
    if (i >= TOTAL) return;

    const float I = (i < HID_N) ? h[i] : o[i - HID_N];

    float v = -65.0f;
    float u = 0.2f * -65.0f;
    v_vec[i] = v;
    u_vec[i] = u;

    for (int s = 1; s < STEPS; ++s) {
        const float vn = v + (0.04f * v * v + 5.0f * v + 0.14f - u + I);
        const float un = u + 0.02f * (0.2f * v - u);
        const bool  sp = (vn >= 30.0f);
        v = sp ? -65.0f : vn;
        u = sp ? (un + 8.0f) : un;
        v_vec[(size_t)s * TOTAL + i] = v;
        u_vec[(size_t)s * TOTAL + i] = u;
    }
}

// ---------------------------------------------------------------------------
// loss = mean((o - target)^2), single workgroup tree reduction.
// ---------------------------------------------------------------------------
__global__ __launch_bounds__(256) void loss_kernel(
    const float* __restrict__ o, const float* __restrict__ tgt,
    float* __restrict__ out)
{
    __shared__ float red[256];
    float acc = 0.f;
    for (int i = threadIdx.x; i < OUT_N; i += 256) {
        const float d = o[i] - tgt[i];
        acc += d * d;
    }
    red[threadIdx.x] = acc;
    __syncthreads();
    for (int s = 128; s > 0; s >>= 1) {
        if (threadIdx.x < s) red[threadIdx.x] += red[threadIdx.x + s];
        __syncthreads();
    }
    if (threadIdx.x == 0) out[0] = red[0] / (float)OUT_N;
}

// ---------------------------------------------------------------------------
// d_in order: v_input, target_output, W1, gain1, fb1, W2, gain2, fb2
// d_out: [ loss(1) | v_vec(100*12288) | u_vec(100*12288) ]
// d_ws:  h(8192 f32) @ 0, o(4096 f32) @ 8192   (48 KB)
// ---------------------------------------------------------------------------
extern "C" void kernel_launch(void* const* d_in, const int* in_sizes, int n_in,
                              void* d_out, int out_size, void* d_ws, size_t ws_size,
                              hipStream_t stream)
{
    const float* v_input = (const float*)d_in[0];
    const float* target  = (const float*)d_in[1];
    const float* W1      = (const float*)d_in[2];
    const float* gain1   = (const float*)d_in[3];
    const float* fb1     = (const float*)d_in[4];
    const float* W2      = (const float*)d_in[5];
    const float* gain2   = (const float*)d_in[6];
    const float* fb2     = (const float*)d_in[7];

    float* h = (float*)d_ws;            // [HID_N]
    float* o = h + HID_N;               // [OUT_N]

    float* out   = (float*)d_out;
    float* loss  = out;                 // [1]
    float* v_vec = out + 1;             // [STEPS*TOTAL]
    float* u_vec = v_vec + (size_t)STEPS * TOTAL;

    // Layer 1: h = sign*log1p(gain1*|(W1-fb1)@x|)
    gemv_wmma_kernel<<<HID_N / 16, 256, 0, stream>>>(W1, fb1, v_input, gain1,
                                                     h, HID_N, IN_N);
    // Layer 2: o = sign*log1p(gain2*|(W2-fb2)@h|)
    gemv_wmma_kernel<<<OUT_N / 16, 256, 0, stream>>>(W2, fb2, h, gain2,
                                                     o, OUT_N, HID_N);
    // Izhikevich dynamics over concat([h, o])
    izhikevich_kernel<<<(TOTAL + 255) / 256, 256, 0, stream>>>(h, o, v_vec, u_vec);
    // MSE loss on layer-2 output
    loss_kernel<<<1, 256, 0, stream>>>(o, target, loss);
}